// GIN_27539330301987
// MI455X (gfx1250) — compile-verified
//
#include <hip/hip_runtime.h>
#include <hip/hip_bf16.h>

typedef __attribute__((ext_vector_type(2))) float v2f;
typedef __attribute__((ext_vector_type(8))) float v8f;

#define FD 64
#define BN_EPS 1e-5f

// ---------------- utility kernels ----------------

__global__ void zero_kernel(float* __restrict__ p, int n) {
    int i = blockIdx.x * blockDim.x + threadIdx.x;
    if (i < n) p[i] = 0.f;
}

__global__ void copy4_kernel(const float4* __restrict__ in, float4* __restrict__ out, long n4) {
    long i = blockIdx.x * (long)blockDim.x + threadIdx.x;
    if (i < n4) out[i] = in[i];
}

// agg[dst[e]][c] += xsrc[src[e]][c] ; one lane per (edge, feature)
__global__ void scatter_add_kernel(const float* __restrict__ xsrc,
                                   const int* __restrict__ src,
                                   const int* __restrict__ dst,
                                   float* __restrict__ agg, int nEdges) {
    long gid = blockIdx.x * (long)blockDim.x + threadIdx.x;
    long e = gid >> 6;                 // FD == 64 features per edge
    int  c = (int)(gid & 63);
    if (e >= nEdges) return;
    int s = src[e];
    int d = dst[e];
    float v = xsrc[(size_t)s * FD + c];
    atomicAdd(agg + (size_t)d * FD + c, v);
}

// ---------------- WMMA f32 GEMM:  out[N,64] = A[N,64] @ W[64,64] + bias ----------------
// One wave handles a 16-row tile. K split into 16 steps of 4 (V_WMMA_F32_16X16X4_F32),
// 4 N-tiles of 16 cols -> 64 WMMA ops per wave.
__global__ void gemm_bias_kernel(const float* __restrict__ A,
                                 const float* __restrict__ W,
                                 const float* __restrict__ bias,
                                 float* __restrict__ out, int nrows) {
    const int lane = threadIdx.x & 31;
    const int wv   = threadIdx.x >> 5;
    const int tile = blockIdx.x * (blockDim.x >> 5) + wv;
    const int ntiles = (nrows + 15) >> 4;
    if (tile >= ntiles) return;            // wave-uniform: EXEC stays all-1s for WMMA

    const int half = lane >> 4;            // 0: K=k..k+1 | 1: K=k+2..k+3  (ISA 16x4 A layout)
    const int l16  = lane & 15;

    int arow = tile * 16 + l16;
    if (arow >= nrows) arow = nrows - 1;   // clamp loads, keep EXEC full
    const float* Ap = A + (size_t)arow * FD;

    v8f c0 = {}, c1 = {}, c2 = {}, c3 = {};

    for (int kk = 0; kk < FD; kk += 4) {
        const int ka = kk + 2 * half;            // K held in VGPR0; VGPR1 holds ka+1
        v2f a = *(const v2f*)(Ap + ka);          // 8B-aligned (ka even)
        const float* Wp = W + (size_t)ka * FD + l16;
        // B 4x16 layout: VGPR v, lane-half h  ->  K = v + 2h, N = l16 (+ 16*nt)
        v2f b0 = { Wp[0],  Wp[FD]      };
        v2f b1 = { Wp[16], Wp[FD + 16] };
        v2f b2 = { Wp[32], Wp[FD + 32] };
        v2f b3 = { Wp[48], Wp[FD + 48] };
        c0 = __builtin_amdgcn_wmma_f32_16x16x4_f32(false, a, false, b0, (short)0, c0, false, false);
        c1 = __builtin_amdgcn_wmma_f32_16x16x4_f32(false, a, false, b1, (short)0, c1, false, false);
        c2 = __builtin_amdgcn_wmma_f32_16x16x4_f32(false, a, false, b2, (short)0, c2, false, false);
        c3 = __builtin_amdgcn_wmma_f32_16x16x4_f32(false, a, false, b3, (short)0, c3, false, false);
    }

    const float bi0 = bias[l16];
    const float bi1 = bias[16 + l16];
    const float bi2 = bias[32 + l16];
    const float bi3 = bias[48 + l16];

    // C/D layout: VGPR g -> row M = g + 8*half, col N = l16 (+16*nt)
    for (int g = 0; g < 8; ++g) {
        int r = tile * 16 + g + 8 * half;
        if (r < nrows) {
            float* op = out + (size_t)r * FD + l16;
            op[0]  = c0[g] + bi0;
            op[16] = c1[g] + bi1;
            op[32] = c2[g] + bi2;
            op[48] = c3[g] + bi3;
        }
    }
}

// ---------------- batchnorm ----------------

__global__ void bn_stats_kernel(const float* __restrict__ h, float* __restrict__ stats, int nrows) {
    __shared__ float s1s[256];
    __shared__ float s2s[256];
    const int col = threadIdx.x & 63;
    const int rep = threadIdx.x >> 6;       // 4 row-replicas per block
    float s1 = 0.f, s2 = 0.f;
    for (int r = blockIdx.x * 4 + rep; r < nrows; r += gridDim.x * 4) {
        float v = h[(size_t)r * FD + col];
        s1 += v;
        s2 += v * v;
    }
    s1s[threadIdx.x] = s1;
    s2s[threadIdx.x] = s2;
    __syncthreads();
    if (threadIdx.x < 64) {
        float a = s1s[threadIdx.x] + s1s[threadIdx.x + 64] + s1s[threadIdx.x + 128] + s1s[threadIdx.x + 192];
        float b = s2s[threadIdx.x] + s2s[threadIdx.x + 64] + s2s[threadIdx.x + 128] + s2s[threadIdx.x + 192];
        atomicAdd(stats + threadIdx.x, a);
        atomicAdd(stats + 64 + threadIdx.x, b);
    }
}

// in-place BN+ReLU on h; also duplicate result into agg (initializes layer-2 aggregation)
__global__ void bn_apply_relu_kernel(float* __restrict__ h, float* __restrict__ agg,
                                     const float* __restrict__ stats,
                                     const float* __restrict__ gamma,
                                     const float* __restrict__ beta, int nrows) {
    long gid = blockIdx.x * (long)blockDim.x + threadIdx.x;
    long n = (long)nrows * FD;
    if (gid >= n) return;
    int col = (int)(gid & 63);
    float invN = 1.f / (float)nrows;
    float mean = stats[col] * invN;
    float var  = stats[64 + col] * invN - mean * mean;
    float v = (h[gid] - mean) * rsqrtf(var + BN_EPS) * gamma[col] + beta[col];
    v = v > 0.f ? v : 0.f;
    h[gid]   = v;
    agg[gid] = v;
}

// ---------------- launch ----------------

extern "C" void kernel_launch(void* const* d_in, const int* in_sizes, int n_in,
                              void* d_out, int out_size, void* d_ws, size_t ws_size,
                              hipStream_t stream) {
    const float* x      = (const float*)d_in[0];
    const int*   ei     = (const int*)  d_in[1];
    const float* W1     = (const float*)d_in[2];
    const float* b1     = (const float*)d_in[3];
    const float* gamma1 = (const float*)d_in[4];
    const float* beta1  = (const float*)d_in[5];
    const float* W2     = (const float*)d_in[6];
    const float* b2     = (const float*)d_in[7];

    const int N = in_sizes[0] / FD;
    const int E = in_sizes[1] / 2;
    const int* src = ei;
    const int* dst = ei + E;

    float* agg   = (float*)d_ws;                 // N*64 floats
    float* h     = agg + (size_t)N * FD;         // N*64 floats
    float* stats = h + (size_t)N * FD;           // 128 floats
    float* out   = (float*)d_out;

    const long nelem = (long)N * FD;
    const long n4 = nelem / 4;
    const long escatter = (long)E * FD;
    const int gemmBlocks = (((N + 15) / 16) + 7) / 8;   // 8 waves per block

    zero_kernel<<<1, 128, 0, stream>>>(stats, 128);

    // ---- layer 1 ----
    copy4_kernel<<<(int)((n4 + 255) / 256), 256, 0, stream>>>((const float4*)x, (float4*)agg, n4);
    scatter_add_kernel<<<(int)((escatter + 255) / 256), 256, 0, stream>>>(x, src, dst, agg, E);
    gemm_bias_kernel<<<gemmBlocks, 256, 0, stream>>>(agg, W1, b1, h, N);
    bn_stats_kernel<<<512, 256, 0, stream>>>(h, stats, N);
    bn_apply_relu_kernel<<<(int)((nelem + 255) / 256), 256, 0, stream>>>(h, agg, stats, gamma1, beta1, N);

    // ---- layer 2 ----
    scatter_add_kernel<<<(int)((escatter + 255) / 256), 256, 0, stream>>>(h, src, dst, agg, E);
    gemm_bias_kernel<<<gemmBlocks, 256, 0, stream>>>(agg, W2, b2, out, N);
}